// HybridQNN_16449724743860
// MI455X (gfx1250) — compile-verified
//
#include <hip/hip_runtime.h>
#include <math.h>

typedef __attribute__((ext_vector_type(2))) float v2f;
typedef __attribute__((ext_vector_type(8))) float v8f;

// Fast tanh: argument range here is a small-weight dot product; formula
// saturates correctly at +-1 for large |x| (exp->inf/0) and propagates NaN.
__device__ __forceinline__ float fast_tanhf(float x) {
  const float e = __expf(2.f * x);                 // v_exp_f32
  return 1.f - 2.f * __builtin_amdgcn_rcpf(e + 1.f); // v_rcp_f32
}

// ---------------------------------------------------------------------------
// Setup kernel (1 block, runs once per launch): build A = Re(U^H Z0 U).
// U = fixed part of circuit (trainable RY/RZ per wire, CNOT ring, final RYs).
// Threads 0..15 each build column k = U e_k in LDS; then 256 threads compute
// A[j][k] = sum_i s_i * Re(conj(U[i,j]) U[i,k]),  s_i = +1 (i<8) / -1 (i>=8).
// Wire w maps to bit (3-w) of the 4-bit state index (wire0 = MSB).
// Uses full-precision sincosf (runs once, feeds all samples).
// ---------------------------------------------------------------------------
__global__ void __launch_bounds__(256) buildA_kernel(const float* __restrict__ qw,
                                                     float* __restrict__ Aout) {
  __shared__ float Ur[16][16];
  __shared__ float Ui[16][16];
  const int t = threadIdx.x;
  if (t < 16) {
    #pragma unroll
    for (int i = 0; i < 16; ++i) { Ur[i][t] = (i == t) ? 1.f : 0.f; Ui[i][t] = 0.f; }
    // trainable RY(qw[0][w]) then RZ(qw[1][w]) per wire
    for (int w = 0; w < 4; ++w) {
      const int mask = 8 >> w;
      float c, s;
      sincosf(0.5f * qw[w], &s, &c);
      #pragma unroll
      for (int i = 0; i < 16; ++i)
        if ((i & mask) == 0) {
          const int j = i | mask;
          float r0 = Ur[i][t], i0 = Ui[i][t], r1 = Ur[j][t], i1 = Ui[j][t];
          Ur[i][t] = c * r0 - s * r1;  Ui[i][t] = c * i0 - s * i1;
          Ur[j][t] = s * r0 + c * r1;  Ui[j][t] = s * i0 + c * i1;
        }
      float cz, sz;
      sincosf(0.5f * qw[4 + w], &sz, &cz);
      #pragma unroll
      for (int i = 0; i < 16; ++i) {
        float re = Ur[i][t], im = Ui[i][t];
        if ((i & mask) == 0) { // * e^{-i th/2}
          Ur[i][t] = re * cz + im * sz;  Ui[i][t] = im * cz - re * sz;
        } else {               // * e^{+i th/2}
          Ur[i][t] = re * cz - im * sz;  Ui[i][t] = im * cz + re * sz;
        }
      }
    }
    // CNOT ring (0,1),(1,2),(2,3),(3,0)
    #pragma unroll
    for (int g = 0; g < 4; ++g) {
      const int mc = 8 >> g, mt = 8 >> ((g + 1) & 3);
      #pragma unroll
      for (int i = 0; i < 16; ++i)
        if ((i & mc) && !(i & mt)) {
          const int j = i | mt;
          float r = Ur[i][t]; Ur[i][t] = Ur[j][t]; Ur[j][t] = r;
          float m = Ui[i][t]; Ui[i][t] = Ui[j][t]; Ui[j][t] = m;
        }
    }
    // final RY(qw[2][w]) per wire
    for (int w = 0; w < 4; ++w) {
      const int mask = 8 >> w;
      float c, s;
      sincosf(0.5f * qw[8 + w], &s, &c);
      #pragma unroll
      for (int i = 0; i < 16; ++i)
        if ((i & mask) == 0) {
          const int j = i | mask;
          float r0 = Ur[i][t], i0 = Ui[i][t], r1 = Ur[j][t], i1 = Ui[j][t];
          Ur[i][t] = c * r0 - s * r1;  Ui[i][t] = c * i0 - s * i1;
          Ur[j][t] = s * r0 + c * r1;  Ui[j][t] = s * i0 + c * i1;
        }
    }
  }
  __syncthreads();
  const int j = t >> 4, k = t & 15;
  float acc = 0.f;
  #pragma unroll
  for (int i = 0; i < 16; ++i) {
    const float sg = (i < 8) ? 1.f : -1.f;
    acc += sg * (Ur[i][j] * Ur[i][k] + Ui[i][j] * Ui[i][k]);
  }
  Aout[j * 16 + k] = acc;
}

// ---------------------------------------------------------------------------
// Main kernel: per wave, tiles of 16 samples.
//   Psi[16x16] (rows = samples, cols = basis),  z = diag(Psi * A * Psi^T)
//   T = Psi*A via 4x V_WMMA_F32_16X16X4_F32, then z_m = sum_n Psi[m,n]*T[m,n]
//   via an in-place LDS product (each P[m,n] owned by exactly one lane;
//   same-wave DS ops are in-order, no barrier needed) + per-lane row sum.
// f32 A-operand layout (ISA 7.12.2): lane l -> row m=l&15, K-pair
// {2*(l>=16), 2*(l>=16)+1}; B-operand mirrored on K; C/D: VGPR v, lanes 0-15
// -> (m=v, n=lane), lanes 16-31 -> (m=v+8, n=lane-16).
// ---------------------------------------------------------------------------
__global__ void __launch_bounds__(256) qnn_main_kernel(
    const float* __restrict__ x,  const float* __restrict__ W1,
    const float* __restrict__ b1, const float* __restrict__ W2,
    const float* __restrict__ b2, const float* __restrict__ W3,
    const float* __restrict__ b3, const float* __restrict__ A,
    float* __restrict__ out, int nsamp) {
  __shared__ float psi_lds[8][16][16];           // per-wave 16x16 Psi tile
  const int lane = threadIdx.x & 31;
  const int wid  = threadIdx.x >> 5;
  float (* __restrict__ psi)[16] = psi_lds[wid];
  const int ml   = lane & 15;                    // row (sample) for A-op
  const int half = lane >> 4;
  const int kh2  = half * 2;                     // K-pair base within K=4 step

  // hoisted tiny weights
  float w1r[4][4], b1r[4];
  #pragma unroll
  for (int i = 0; i < 4; ++i) {
    b1r[i] = b1[i];
    #pragma unroll
    for (int j = 0; j < 4; ++j) w1r[i][j] = W1[i * 4 + j];
  }
  float w2r[4], b2r[4], w30[4], w31[4];
  #pragma unroll
  for (int j = 0; j < 4; ++j) {
    w2r[j] = W2[j]; b2r[j] = b2[j]; w30[j] = W3[j]; w31[j] = W3[4 + j];
  }
  const float b30 = b3[0], b31 = b3[1];

  // hoisted B operand: bop[t] = rows (4t+kh2, 4t+kh2+1) of A at column n=ml
  v2f bop[4];
  #pragma unroll
  for (int st = 0; st < 4; ++st) {
    bop[st].x = A[(4 * st + kh2) * 16 + ml];
    bop[st].y = A[(4 * st + kh2 + 1) * 16 + ml];
  }

  const int nt = nsamp >> 4;                     // 16-sample tiles
  const int gw = blockIdx.x * (blockDim.x >> 5) + wid;
  const int nw = gridDim.x * (blockDim.x >> 5);

  for (int tile = gw; tile < nt; tile += nw) {
    const int s = tile * 16 + ml;
    if (tile + nw < nt)
      __builtin_prefetch(&x[(size_t)(s + nw * 16) * 4], 0, 3);

    const float4 xr = ((const float4*)x)[s];

    // angles -> cos/sin of h/2 for this lane's sample. h/2 in (-0.5, 0.5) rad
    // so hardware v_sin/v_cos need no range reduction.
    float cth[4], sth[4];
    #pragma unroll
    for (int i = 0; i < 4; ++i) {
      const float h = fast_tanhf(xr.x * w1r[i][0] + xr.y * w1r[i][1] +
                                 xr.z * w1r[i][2] + xr.w * w1r[i][3] + b1r[i]);
      const float a = 0.5f * h;
      sth[i] = __sinf(a);                        // v_sin_f32
      cth[i] = __cosf(a);                        // v_cos_f32
    }
    // psi_n = (n&8?s0:c0)(n&4?s1:c1)(n&2?s2:c2)(n&1?s3:c3); this lane needs
    // n = 4t+kh2 (+1): bit8 = t>=2, bit4 = t&1, bit2 = half, bit1 = 0/1.
    const float f2 = half ? sth[2] : cth[2];
    v2f aop[4];
    #pragma unroll
    for (int st = 0; st < 4; ++st) {
      const float f0 = (st >= 2) ? sth[0] : cth[0];
      const float f1 = (st & 1) ? sth[1] : cth[1];
      const float base = f0 * f1 * f2;
      aop[st].x = base * cth[3];
      aop[st].y = base * sth[3];
      psi[ml][4 * st + kh2]     = aop[st].x;     // stage Psi for C-layout use
      psi[ml][4 * st + kh2 + 1] = aop[st].y;
    }

    // T = Psi * A : accumulate 4 K=4 slices
    v8f d = {0.f, 0.f, 0.f, 0.f, 0.f, 0.f, 0.f, 0.f};
    #pragma unroll
    for (int st = 0; st < 4; ++st)
      d = __builtin_amdgcn_wmma_f32_16x16x4_f32(
              false, aop[st], false, bop[st], (short)0, d, false, false);

    // In-place product: P[m,n] = Psi[m,n] * T[m,n]. Address (v+8*half, ml) is
    // touched by exactly one lane -> no cross-lane hazard; same-wave DS ops
    // are processed in order, so the row reads below see all 32 lanes' writes.
    #pragma unroll
    for (int v = 0; v < 8; ++v)
      psi[v + 8 * half][ml] *= d[v];

    // z_m = row sum; every lane sums row ml (lanes 16-31 duplicate, harmless)
    const float4 r0 = *reinterpret_cast<const float4*>(&psi[ml][0]);
    const float4 r1 = *reinterpret_cast<const float4*>(&psi[ml][4]);
    const float4 r2 = *reinterpret_cast<const float4*>(&psi[ml][8]);
    const float4 r3 = *reinterpret_cast<const float4*>(&psi[ml][12]);
    const float z = ((r0.x + r0.y) + (r0.z + r0.w)) +
                    ((r1.x + r1.y) + (r1.z + r1.w)) +
                    ((r2.x + r2.y) + (r2.z + r2.w)) +
                    ((r3.x + r3.y) + (r3.z + r3.w));

    if (lane < 16) {                             // lanes 0..15 own samples
      float o0 = b30, o1 = b31;
      #pragma unroll
      for (int j = 0; j < 4; ++j) {
        const float h2 = fmaxf(0.f, z * w2r[j] + b2r[j]);
        o0 += w30[j] * h2;
        o1 += w31[j] * h2;
      }
      ((float2*)out)[s] = make_float2(o0, o1);
    }
  }
}

extern "C" void kernel_launch(void* const* d_in, const int* in_sizes, int n_in,
                              void* d_out, int out_size, void* d_ws, size_t ws_size,
                              hipStream_t stream) {
  const float* x  = (const float*)d_in[0];
  const float* W1 = (const float*)d_in[1];
  const float* b1 = (const float*)d_in[2];
  const float* qw = (const float*)d_in[3];
  const float* W2 = (const float*)d_in[4];
  const float* b2 = (const float*)d_in[5];
  const float* W3 = (const float*)d_in[6];
  const float* b3 = (const float*)d_in[7];
  float* A = (float*)d_ws;                       // 256 floats
  float* out = (float*)d_out;

  const int nsamp = in_sizes[0] / 4;             // B
  buildA_kernel<<<1, 256, 0, stream>>>(qw, A);

  const int nt = nsamp >> 4;
  int blocks = (nt + 7) / 8;                     // 8 waves per block
  if (blocks > 2048) blocks = 2048;
  if (blocks < 1) blocks = 1;
  qnn_main_kernel<<<blocks, 256, 0, stream>>>(x, W1, b1, W2, b2, W3, b3, A,
                                              out, nsamp);
}